// WeightedPairwiseLoss_47399259079022
// MI455X (gfx1250) — compile-verified
//
#include <hip/hip_runtime.h>
#include <hip/hip_bf16.h>

// ---------------------------------------------------------------------------
// WeightedPairwiseLoss on MI455X (gfx1250, wave32)
//   B=32, N=4096, K=819. Compute-bound on TRANS pipe (~43M exp/log).
//   CDNA5 paths: v_wmma_f32_16x16x4_f32 wave reductions,
//                global_load_async_to_lds_b32 LDS staging (ASYNCcnt),
//                global_prefetch_b8.
// ---------------------------------------------------------------------------

typedef __attribute__((ext_vector_type(2))) float v2f;
typedef __attribute__((ext_vector_type(8))) float v8f;

constexpr int   B_   = 32;
constexpr int   N_   = 4096;
constexpr int   K_   = 819;          // int(4096 * 0.2)
constexpr float EPS_  = 1e-8f;
constexpr float LAMB_ = 0.25f;
constexpr float LOGC_ = -100.0f;

// workspace layout (float offsets)
constexpr int OFF_NUM  = 0;                      // [32] pairwise numerators
constexpr int OFF_SA   = 32;                     // [32] sum sqrt(w) over top
constexpr int OFF_SB   = 64;                     // [32] sum sqrt(w) over bot
constexpr int OFF_BCEN = 96;                     // [32] sum bce*w*m
constexpr int OFF_WM   = 128;                    // [32] sum w*m
constexpr int OFF_P    = 160;                    // sum p*m (global)
constexpr int OFF_M    = 161;                    // sum m   (global)
constexpr int OFF_TOPS = 192;                    // [B*K] top scores
constexpr int OFF_TOPA = OFF_TOPS + B_ * K_;     // [B*K] top sqrt(w)
constexpr int OFF_BOTS = OFF_TOPA + B_ * K_;     // [B*K] bot scores
constexpr int OFF_BOTB = OFF_BOTS + B_ * K_;     // [B*K] bot sqrt(w)

// --- CDNA5 async global->LDS staging (ASYNCcnt path) -----------------------
// Flat addresses of __shared__ objects carry the LDS byte offset in their low
// 32 bits (ISA 10.2), which is exactly what the
// VDST lds-address operand of GLOBAL_LOAD_ASYNC_TO_LDS_B32 wants.
__device__ __forceinline__ void async_copy_f32(float* lds_dst, const float* g_src) {
  unsigned int       la = (unsigned int)(uintptr_t)lds_dst;
  unsigned long long ga = (unsigned long long)(uintptr_t)g_src;
  asm volatile("global_load_async_to_lds_b32 %0, %1, off"
               :: "v"(la), "v"(ga) : "memory");
}
__device__ __forceinline__ void async_wait0() {
  asm volatile("s_wait_asynccnt 0x0" ::: "memory");
}

// --- wave32 reduction via one 16x16x4 F32 WMMA -----------------------------
// B operand (4x16) carries the 32 lane-partials (b.y = 0); A = ones(16x4).
// D_{r,c} = sum_k B_{k,c}: each B element lands in exactly one column sum
// regardless of the lane<->(k,c) mapping, so folding the 16 column sums
// (d[0]: lanes 0..15 = row M=0, lanes 16..31 duplicate as row M=8) gives the
// exact total of all 32 inputs.
__device__ __forceinline__ float wave_reduce_wmma(float v) {
  v2f a; a.x = 1.0f; a.y = 1.0f;       // A: 16x4 all-ones
  v2f b; b.x = v;    b.y = 0.0f;       // B: 4x16 carrying the 32 partials
  v8f c = {};
  v8f d = __builtin_amdgcn_wmma_f32_16x16x4_f32(
      /*neg_a=*/false, a, /*neg_b=*/false, b,
      /*c_mod=*/(short)0, c, /*reuse_a=*/false, /*reuse_b=*/false);
  float s = d[0];
  for (int off = 8; off >= 1; off >>= 1) s += __shfl_xor(s, off, 32);
  return s;                             // total on every lane
}

// valid on thread 0; must be called by ALL threads of the block (EXEC all-1s)
__device__ __forceinline__ float block_reduce(float v, float* red) {
  __syncthreads();                      // protect red across successive calls
  float wsum = wave_reduce_wmma(v);
  const int lane = threadIdx.x & 31;
  const int wv   = threadIdx.x >> 5;
  if (lane == 0) red[wv] = wsum;
  __syncthreads();
  float r = 0.0f;
  if (threadIdx.x == 0) {
    const int nw = (int)(blockDim.x >> 5);
    for (int i = 0; i < nw; ++i) r += red[i];
  }
  return r;
}

// stable softplus: max(x,0) + log(1 + exp(-|x|)); argument of the log is in
// (1,2], so the direct v_log_f32 form matches log1p at fp32 accuracy while
// costing 2 TRANS + a few VALU (TRANS co-executes with VALU on CDNA5).
__device__ __forceinline__ float softplusf(float x) {
  float e = __expf(-fabsf(x));
  return fmaxf(x, 0.0f) + __logf(1.0f + e);
}

// --- K0: zero accumulators --------------------------------------------------
__global__ void wpl_init_kernel(float* __restrict__ ws) {
  int t = threadIdx.x;
  if (t < OFF_TOPS) ws[t] = 0.0f;
}

// --- K1: per-row bitonic value sort -> thresholds -> compaction -------------
__global__ void __launch_bounds__(512)
wpl_select_kernel(const float* __restrict__ yrank,
                  const float* __restrict__ scores,
                  const float* __restrict__ weights,
                  float* __restrict__ ws) {
  const int b   = blockIdx.x;
  const int tid = threadIdx.x;
  constexpr int TB = 512;
  __shared__ float vals[N_];            // 16 KB of 320 KB LDS
  __shared__ float red[16];
  __shared__ int   cTop, cBot;
  __shared__ float thrTop, thrBot;

  const size_t base = (size_t)b * N_;
  // async-stage the whole y_rank row into LDS (no VGPR round trip)
  for (int e = tid; e < N_; e += TB) async_copy_f32(&vals[e], &yrank[base + e]);
  async_wait0();
  __syncthreads();

  // bitonic sort (ascending), values only — the pair sum is order-invariant
  for (int size = 2; size <= N_; size <<= 1) {
    for (int stride = size >> 1; stride > 0; stride >>= 1) {
      for (int t = tid; t < N_ / 2; t += TB) {
        int lo = ((t / stride) * (stride << 1)) + (t % stride);
        int hi = lo + stride;
        bool asc = ((lo & size) == 0);
        float x = vals[lo], y = vals[hi];
        if ((x > y) == asc) { vals[lo] = y; vals[hi] = x; }
      }
      __syncthreads();
    }
  }
  if (tid == 0) {
    thrBot = vals[K_ - 1];              // k-th smallest
    thrTop = vals[N_ - K_];             // k-th largest
    cTop = 0; cBot = 0;
  }
  __syncthreads();

  float* __restrict__ top_s = ws + OFF_TOPS + (size_t)b * K_;
  float* __restrict__ top_a = ws + OFF_TOPA + (size_t)b * K_;
  float* __restrict__ bot_s = ws + OFF_BOTS + (size_t)b * K_;
  float* __restrict__ bot_b = ws + OFF_BOTB + (size_t)b * K_;

  float sA = 0.0f, sB = 0.0f;
  for (int e = tid; e < N_; e += TB) {
    float yv = yrank[base + e];
    if (yv >= thrTop) {
      int pos = atomicAdd(&cTop, 1);
      if (pos < K_) {
        float sq = sqrtf(weights[base + e]);
        top_s[pos] = scores[base + e];
        top_a[pos] = sq;
        sA += sq;
      }
    }
    if (yv <= thrBot) {
      int pos = atomicAdd(&cBot, 1);
      if (pos < K_) {
        float sq = sqrtf(weights[base + e]);
        bot_s[pos] = scores[base + e];
        bot_b[pos] = sq;
        sB += sq;
      }
    }
  }
  float tA = block_reduce(sA, red);
  float tB = block_reduce(sB, red);
  if (tid == 0) { ws[OFF_SA + b] = tA; ws[OFF_SB + b] = tB; }
}

// --- K2: pairwise softplus numerator ---------------------------------------
// grid (32 rows, 4 i-tiles of 256, 8 j-slices); thread owns one i.
constexpr int JSL_  = 8;
constexpr int JCH_  = (K_ + JSL_ - 1) / JSL_;     // 103

__global__ void __launch_bounds__(256)
wpl_pair_kernel(float* __restrict__ ws) {
  const int b   = blockIdx.x;
  const int it  = blockIdx.y;
  const int js  = blockIdx.z;
  const int tid = threadIdx.x;
  __shared__ float s_bot[JCH_];
  __shared__ float w_bot[JCH_];
  __shared__ float red[8];

  const float* __restrict__ top_s = ws + OFF_TOPS + (size_t)b * K_;
  const float* __restrict__ top_a = ws + OFF_TOPA + (size_t)b * K_;
  const float* __restrict__ bot_s = ws + OFF_BOTS + (size_t)b * K_;
  const float* __restrict__ bot_b = ws + OFF_BOTB + (size_t)b * K_;

  const int i = it * 256 + tid;
  // CDNA5 global_prefetch_b8 of the top tile while the bot slice streams in
  __builtin_prefetch(top_s + i, 0, 0);
  __builtin_prefetch(top_a + i, 0, 0);

  const int jstart = js * JCH_;
  const int jn     = min(K_ - jstart, JCH_);
  // async-stage the bot slice into LDS on the ASYNCcnt path
  for (int t = tid; t < jn; t += 256) {
    async_copy_f32(&s_bot[t], &bot_s[jstart + t]);
    async_copy_f32(&w_bot[t], &bot_b[jstart + t]);
  }
  async_wait0();
  __syncthreads();

  float partial = 0.0f;
  if (i < K_) {
    const float s_i = top_s[i];
    const float a_i = top_a[i];
    float acc = 0.0f;
#pragma unroll 4
    for (int j = 0; j < jn; ++j) {
      // loss_ij = softplus(-(s_i - s_j)) = softplus(s_j - s_i)
      acc = fmaf(w_bot[j], softplusf(s_bot[j] - s_i), acc);
    }
    partial = a_i * acc;
  }
  float tot = block_reduce(partial, red);
  if (tid == 0) atomicAdd(&ws[OFF_NUM + b], tot);
}

// --- K3: BCE trade loss + mean_p sums --------------------------------------
__global__ void __launch_bounds__(256)
wpl_bce_kernel(const float* __restrict__ p_trade,
               const float* __restrict__ y_trade,
               const float* __restrict__ weights,
               const unsigned char* __restrict__ mask,
               float* __restrict__ ws) {
  const int b   = blockIdx.x;
  const int tid = threadIdx.x;
  __shared__ float red[8];
  const size_t base = (size_t)b * N_;
  float num = 0.0f, den = 0.0f, ps = 0.0f, ms = 0.0f;
  for (int e = tid; e < N_; e += 256) {
    float pv = p_trade[base + e];
    float yv = y_trade[base + e];
    float wv = weights[base + e];
    float m  = mask[base + e] ? 1.0f : 0.0f;
    // p in (1e-4, 1-1e-4): direct logs are exact enough; clamp per reference
    float lp  = fmaxf(__logf(pv),        LOGC_);
    float l1  = fmaxf(__logf(1.0f - pv), LOGC_);
    float bce = -(yv * lp + (1.0f - yv) * l1);
    float wm  = wv * m;
    num = fmaf(bce, wm, num);
    den += wm;
    ps  = fmaf(pv, m, ps);
    ms  += m;
  }
  float tnum = block_reduce(num, red);
  float tden = block_reduce(den, red);
  float tps  = block_reduce(ps,  red);
  float tms  = block_reduce(ms,  red);
  if (tid == 0) {
    ws[OFF_BCEN + b] = tnum;
    ws[OFF_WM   + b] = tden;
    atomicAdd(&ws[OFF_P], tps);
    atomicAdd(&ws[OFF_M], tms);
  }
}

// --- K4: combine -----------------------------------------------------------
__global__ void wpl_final_kernel(const float* __restrict__ ws,
                                 float* __restrict__ out) {
  const int b = threadIdx.x;            // 32 threads = one wave = one row each
  float lr = 0.0f, lt = 0.0f;
  if (b < B_) {
    float denom = ws[OFF_SA + b] * ws[OFF_SB + b] + EPS_;   // separable!
    lr = ws[OFF_NUM + b] / denom;
    lt = ws[OFF_BCEN + b] / (ws[OFF_WM + b] + EPS_);
  }
  for (int off = 16; off >= 1; off >>= 1) {
    lr += __shfl_xor(lr, off, 32);
    lt += __shfl_xor(lt, off, 32);
  }
  if (b == 0) {
    float avg_rank  = lr / (float)B_;
    float avg_trade = lt / (float)B_;
    out[0] = avg_rank + LAMB_ * avg_trade;  // total
    out[1] = avg_rank;
    out[2] = avg_trade;
    out[3] = ws[OFF_P] / ws[OFF_M];         // mean_p_trade
  }
}

// ---------------------------------------------------------------------------
extern "C" void kernel_launch(void* const* d_in, const int* in_sizes, int n_in,
                              void* d_out, int out_size, void* d_ws, size_t ws_size,
                              hipStream_t stream) {
  const float*         scores  = (const float*)d_in[0];
  const float*         p_trade = (const float*)d_in[1];
  const float*         y_rank  = (const float*)d_in[2];
  const float*         y_trade = (const float*)d_in[3];
  const float*         weights = (const float*)d_in[4];
  const unsigned char* mask    = (const unsigned char*)d_in[5];
  float* ws  = (float*)d_ws;
  float* out = (float*)d_out;

  wpl_init_kernel<<<1, 256, 0, stream>>>(ws);
  wpl_select_kernel<<<dim3(B_), 512, 0, stream>>>(y_rank, scores, weights, ws);
  wpl_pair_kernel<<<dim3(B_, 4, JSL_), 256, 0, stream>>>(ws);
  wpl_bce_kernel<<<dim3(B_), 256, 0, stream>>>(p_trade, y_trade, weights, mask, ws);
  wpl_final_kernel<<<1, 32, 0, stream>>>(ws, out);
}